// ResGCN_17480516895406
// MI455X (gfx1250) — compile-verified
//
#include <hip/hip_runtime.h>
#include <hip/hip_bf16.h>
#include <math.h>

// ---------------- problem constants ----------------
#define NN      1024      // nodes
#define IN_DIM  16
#define HID     32
#define OUT_DIM 256
#define NE      8192      // edges (without self loops)
#define FDIM    32768     // N*HID
#define F4      8192      // FDIM/4
#define F8      4096      // FDIM/8
#define CHUNK   512       // GEMV reduction chunk (rows)

typedef __attribute__((ext_vector_type(2))) float v2f;
typedef __attribute__((ext_vector_type(8))) float v8f;

// =====================================================================
// Graph preprocessing (deterministic CSR build, no float atomics)
// =====================================================================

// deg[n] = 1 (self loop) + #edges with col==n ; dinv = rsqrt(deg)
__global__ void k_build_deg(const int* __restrict__ col, int* __restrict__ cnt,
                            float* __restrict__ dinv) {
    int n = blockIdx.x * blockDim.x + threadIdx.x;
    if (n >= NN) return;
    int c = 0;
    for (int e = 0; e < NE; ++e) c += (col[e] == n) ? 1 : 0;
    cnt[n] = c;
    dinv[n] = rsqrtf((float)(c + 1));
}

// exclusive prefix scan of cnt[1024] -> offs[1025]  (single block)
__global__ void k_scan(const int* __restrict__ cnt, int* __restrict__ offs) {
    __shared__ int s[NN];
    int t = threadIdx.x;
    s[t] = cnt[t];
    __syncthreads();
    for (int d = 1; d < NN; d <<= 1) {
        int v = (t >= d) ? s[t - d] : 0;
        __syncthreads();
        s[t] += v;
        __syncthreads();
    }
    offs[t + 1] = s[t];
    if (t == 0) offs[0] = 0;
}

// each node scans edges in order and writes its bucket -> deterministic order
__global__ void k_fill_csr(const int* __restrict__ col, const int* __restrict__ offs,
                           int* __restrict__ csr) {
    int n = blockIdx.x * blockDim.x + threadIdx.x;
    if (n >= NN) return;
    int p = offs[n];
    for (int e = 0; e < NE; ++e)
        if (col[e] == n) csr[p++] = e;
}

// =====================================================================
// Dense GEMM [1024,K] @ [K,Ncols] via V_WMMA_F32_16X16X4_F32.
// One wave (32 threads) per 16x16 output tile. EXEC is all-ones.
// A fragment (16x4 f32): lane m = lane&15, K-pair = (lane>>4)*2.
// B fragment (4x16 f32): lane n = lane&15, K-pair = (lane>>4)*2.
// C/D (16x16 f32): VGPR r -> M = r + 8*(lane>>4), N = lane&15.
// =====================================================================
__global__ void k_gemm_wmma(const float* __restrict__ A, const float* __restrict__ B,
                            float* __restrict__ C, int K, int Ncols) {
    const int tm   = blockIdx.x << 4;
    const int tn   = blockIdx.y << 4;
    const int lane = threadIdx.x;      // 0..31
    const int half = lane >> 4;        // 0|1
    const int l16  = lane & 15;

    v8f acc = {};
    for (int k0 = 0; k0 < K; k0 += 4) {
        const int ka = k0 + half * 2;
        v2f a, b;
        a.x = A[(size_t)(tm + l16) * K + ka];
        a.y = A[(size_t)(tm + l16) * K + ka + 1];
        b.x = B[(size_t)(ka)     * Ncols + tn + l16];
        b.y = B[(size_t)(ka + 1) * Ncols + tn + l16];
        acc = __builtin_amdgcn_wmma_f32_16x16x4_f32(
            /*neg_a=*/false, a, /*neg_b=*/false, b,
            /*c_mod=*/(short)0, acc, /*reuse_a=*/false, /*reuse_b=*/false);
    }
#pragma unroll
    for (int r = 0; r < 8; ++r) {
        int m = r + half * 8;
        C[(size_t)(tm + m) * Ncols + tn + l16] = acc[r];
    }
}

// =====================================================================
// GCN aggregation: out[c,f] = act( dinv[c]*( dinv[c]*hw[c,f]
//                      + sum_{e in csr[c]} dinv[row[e]]*hw[row[e],f] )
//                      + bias[f] (+ identity[c,f]) )
// one node per block (32 threads = 32 features). Deterministic order.
// =====================================================================
__global__ void k_aggregate(const float* __restrict__ hw, const float* __restrict__ dinv,
                            const int* __restrict__ offs, const int* __restrict__ csr,
                            const int* __restrict__ row, const float* __restrict__ bias,
                            const float* __restrict__ identity, float* __restrict__ out,
                            int do_relu) {
    int c = blockIdx.x;
    int f = threadIdx.x;
    float dc  = dinv[c];
    float acc = dc * hw[(size_t)c * HID + f];        // self loop
    int beg = offs[c], end = offs[c + 1];
    for (int j = beg; j < end; ++j) {
        int e = csr[j];
        int r = row[e];
        acc = fmaf(dinv[r], hw[(size_t)r * HID + f], acc);
    }
    float v = fmaf(dc, acc, bias[f]);
    if (identity) v += identity[(size_t)c * HID + f];
    if (do_relu)  v = fmaxf(v, 0.0f);
    out[(size_t)c * HID + f] = v;
}

// =====================================================================
// Streaming GEMV (HBM-bound): each thread owns a float4 column group,
// rows split into CHUNK-sized reduction chunks (grid.y). Row-major W ->
// global_load_b128 fully coalesced; v-chunk staged in LDS; prefetch
// ahead (global_prefetch_b8) to keep the HBM pipeline primed.
// partial[chunk, j] is reduced deterministically in a second pass.
// =====================================================================
__global__ void k_gemv_partial4(const float* __restrict__ v, const float* __restrict__ W,
                                float* __restrict__ partial, int n_out) {
    const int jv = blockIdx.x * blockDim.x + threadIdx.x;   // float4 column index
    const int c  = blockIdx.y;                              // row chunk
    const int i0 = c * CHUNK;

    __shared__ float sv[CHUNK];
    for (int t = threadIdx.x; t < CHUNK; t += blockDim.x) sv[t] = v[i0 + t];
    __syncthreads();

    const int     s4 = n_out >> 2;                          // row stride in float4s
    const float4* Wp = (const float4*)(W + (size_t)i0 * n_out) + jv;
    float4 acc = make_float4(0.f, 0.f, 0.f, 0.f);
#pragma unroll 4
    for (int i = 0; i < CHUNK; ++i) {
        __builtin_prefetch((const void*)(Wp + (size_t)(i + 16) * s4), 0, 0);
        float4 w = Wp[(size_t)i * s4];
        float  s = sv[i];
        acc.x = fmaf(s, w.x, acc.x);
        acc.y = fmaf(s, w.y, acc.y);
        acc.z = fmaf(s, w.z, acc.z);
        acc.w = fmaf(s, w.w, acc.w);
    }
    ((float4*)(partial + (size_t)c * n_out))[jv] = acc;
}

__global__ void k_reduce_bias_act(const float* __restrict__ partial,
                                  const float* __restrict__ bias,
                                  float* __restrict__ out,
                                  int n_out, int n_chunks, int do_relu) {
    int j = blockIdx.x * 256 + threadIdx.x;
    if (j >= n_out) return;
    float acc = bias[j];
    for (int c = 0; c < n_chunks; ++c) acc += partial[(size_t)c * n_out + j];
    out[j] = do_relu ? fmaxf(acc, 0.0f) : acc;
}

// =====================================================================
// Softmax over 256 logits (single block).
// =====================================================================
__global__ void k_softmax256(const float* __restrict__ logits, float* __restrict__ out) {
    __shared__ float s[OUT_DIM];
    int t = threadIdx.x;
    float x = logits[t];
    s[t] = x; __syncthreads();
    for (int d = 128; d > 0; d >>= 1) {
        if (t < d) s[t] = fmaxf(s[t], s[t + d]);
        __syncthreads();
    }
    float m = s[0]; __syncthreads();
    float e = expf(x - m);
    s[t] = e; __syncthreads();
    for (int d = 128; d > 0; d >>= 1) {
        if (t < d) s[t] += s[t + d];
        __syncthreads();
    }
    out[t] = e / s[0];
}

// =====================================================================
// host-side orchestration
// =====================================================================
extern "C" void kernel_launch(void* const* d_in, const int* in_sizes, int n_in,
                              void* d_out, int out_size, void* d_ws, size_t ws_size,
                              hipStream_t stream) {
    const float* x    = (const float*)d_in[0];
    const int*   ei   = (const int*)  d_in[1];   // [2, NE]
    const float* W_in = (const float*)d_in[2];
    const float* b_in = (const float*)d_in[3];
    const float* Wb   = (const float*)d_in[4];   // [3,2,32,32]
    const float* bb   = (const float*)d_in[5];   // [3,2,32]
    const float* W1   = (const float*)d_in[6];
    const float* b1   = (const float*)d_in[7];
    const float* W2   = (const float*)d_in[8];
    const float* b2   = (const float*)d_in[9];
    const float* W3   = (const float*)d_in[10];
    const float* b3   = (const float*)d_in[11];
    float* out = (float*)d_out;

    const int* row = ei;
    const int* col = ei + NE;

    // ---- workspace carving (256B aligned) ----
    char* base = (char*)d_ws;
    size_t off = 0;
    auto carve = [&](size_t bytes) -> char* {
        char* p = base + off;
        off = (off + bytes + 255) & ~(size_t)255;
        return p;
    };
    int*   cnt   = (int*)  carve(NN * 4);
    int*   offs  = (int*)  carve((NN + 1) * 4);
    int*   csr   = (int*)  carve(NE * 4);
    float* dinv  = (float*)carve(NN * 4);
    float* hw    = (float*)carve(FDIM * 4);
    float* h     = (float*)carve(FDIM * 4);
    float* o     = (float*)carve(FDIM * 4);
    float* v1    = (float*)carve(F4 * 4);
    float* v2    = (float*)carve(F8 * 4);
    float* logit = (float*)carve(OUT_DIM * 4);
    float* p1    = (float*)carve((size_t)(FDIM / CHUNK) * F4 * 4);   // 64*8192*4 = 2 MB
    float* p2    = (float*)carve((size_t)(F4 / CHUNK)   * F8 * 4);   // 16*4096*4
    float* p3    = (float*)carve((size_t)(F8 / CHUNK)   * OUT_DIM * 4);
    (void)ws_size; (void)n_in; (void)in_sizes; (void)out_size;

    // ---- graph preprocessing ----
    k_build_deg<<<NN / 256, 256, 0, stream>>>(col, cnt, dinv);
    k_scan     <<<1, NN, 0, stream>>>(cnt, offs);
    k_fill_csr <<<NN / 256, 256, 0, stream>>>(col, offs, csr);

    dim3 wave(32, 1, 1);

    // ---- input conv: h = relu(agg(x @ W_in) + b_in) ----
    k_gemm_wmma<<<dim3(NN / 16, HID / 16), wave, 0, stream>>>(x, W_in, hw, IN_DIM, HID);
    k_aggregate<<<NN, HID, 0, stream>>>(hw, dinv, offs, csr, row, b_in, nullptr, h, 1);

    // ---- residual blocks ----
    for (int blk = 0; blk < 3; ++blk) {
        const float* W0  = Wb + (size_t)(blk * 2 + 0) * HID * HID;
        const float* W1b = Wb + (size_t)(blk * 2 + 1) * HID * HID;
        const float* bb0 = bb + (size_t)(blk * 2 + 0) * HID;
        const float* bb1 = bb + (size_t)(blk * 2 + 1) * HID;
        // o = relu(conv(h, W0))
        k_gemm_wmma<<<dim3(NN / 16, HID / 16), wave, 0, stream>>>(h, W0, hw, HID, HID);
        k_aggregate<<<NN, HID, 0, stream>>>(hw, dinv, offs, csr, row, bb0, nullptr, o, 1);
        // h = relu(conv(o, W1b) + h)   (identity folded; per-element in-place safe)
        k_gemm_wmma<<<dim3(NN / 16, HID / 16), wave, 0, stream>>>(o, W1b, hw, HID, HID);
        k_aggregate<<<NN, HID, 0, stream>>>(hw, dinv, offs, csr, row, bb1, h, h, 1);
    }

    // ---- MLP head (HBM-bound streaming GEMVs, deterministic 2-pass) ----
    // layer 1: [32768] @ [32768,8192]: 64 chunks x 8 col-blocks = 512 WGs, b128 streams
    k_gemv_partial4<<<dim3(F4 / 4 / 256, FDIM / CHUNK), 256, 0, stream>>>(h, W1, p1, F4);
    k_reduce_bias_act<<<F4 / 256, 256, 0, stream>>>(p1, b1, v1, F4, FDIM / CHUNK, 1);
    // layer 2: [8192] @ [8192,4096]
    k_gemv_partial4<<<dim3(F8 / 4 / 256, F4 / CHUNK), 256, 0, stream>>>(v1, W2, p2, F8);
    k_reduce_bias_act<<<F8 / 256, 256, 0, stream>>>(p2, b2, v2, F8, F4 / CHUNK, 1);
    // layer 3: [4096] @ [4096,256]  (64 threads x 4 cols = 256 outputs)
    k_gemv_partial4<<<dim3(1, F8 / CHUNK), 64, 0, stream>>>(v2, W3, p3, OUT_DIM);
    k_reduce_bias_act<<<1, 256, 0, stream>>>(p3, b3, logit, OUT_DIM, F8 / CHUNK, 0);

    // ---- softmax ----
    k_softmax256<<<1, OUT_DIM, 0, stream>>>(logit, out);
}